// TrendAttentionLayer_47132971107264
// MI455X (gfx1250) — compile-verified
//
#include <hip/hip_runtime.h>
#include <hip/hip_bf16.h>
#include <math.h>

typedef __attribute__((ext_vector_type(2))) float v2f;
typedef __attribute__((ext_vector_type(8))) float v8f;

#define D_MODEL 512
#define L_SEQ   2048
#define N_HEADS 8
#define E_HEAD  64
#define LDS_STRIDE 520   // even -> 8B-aligned v2f LDS loads, avoids bank-conflict pathologies

// ---------------------------------------------------------------------------
// Kernel 1: fused projection  Out(4096x512) = X(4096x512) @ W(512x512) + b
// One wave -> one 16x16 tile, K-loop of 128 x V_WMMA_F32_16X16X4_F32.
// W column panel staged in LDS (transposed) so B fragments are ds_load_b64.
// ---------------------------------------------------------------------------
__launch_bounds__(256)
__global__ void qkv_proj_kernel(const float* __restrict__ X,
                                const float* __restrict__ W,
                                const float* __restrict__ bias,
                                float* __restrict__ Out) {
    alignas(16) __shared__ float wpanel[16 * LDS_STRIDE];
    const int tid  = threadIdx.x;
    const int wave = tid >> 5;
    const int lane = tid & 31;
    const int n0   = blockIdx.x * 16;
    const int row0 = blockIdx.y * 128 + wave * 16;

    // Cooperative load of W[0:512, n0:n0+16] into LDS as wpanel[n][k]
    for (int idx = tid; idx < D_MODEL * 16; idx += 256) {
        int k = idx >> 4;
        int n = idx & 15;
        wpanel[n * LDS_STRIDE + k] = W[(size_t)k * D_MODEL + n0 + n];
    }
    __syncthreads();

    const int m  = lane & 15;            // A row / B col / C col
    const int kh = (lane >> 4) << 1;     // K sub-offset: 0 or 2
    const float* arow = X + (size_t)(row0 + m) * D_MODEL;
    const float* brow = wpanel + m * LDS_STRIDE;

    v8f acc = {};
    #pragma unroll 4
    for (int kb = 0; kb < D_MODEL; kb += 4) {
        v2f a = *(const v2f*)(arow + kb + kh);
        v2f b = *(const v2f*)(brow + kb + kh);
        acc = __builtin_amdgcn_wmma_f32_16x16x4_f32(false, a, false, b,
                                                    (short)0, acc, false, false);
    }

    const float bn   = bias[n0 + m];
    const int   mrow = row0 + ((lane >> 4) << 3);   // +8 rows for upper lane half
    #pragma unroll
    for (int r = 0; r < 8; ++r)
        Out[(size_t)(mrow + r) * D_MODEL + n0 + m] = acc[r] + bn;
}

// ---------------------------------------------------------------------------
// Kernel 2: attention GEMM per (b,h):  attn1 = Q(2048x64) @ K^T(64x2048)
// One wave -> 16x64 output (4 accumulators), A fragment reused 4x.
// ---------------------------------------------------------------------------
__launch_bounds__(256)
__global__ void attn_gemm_kernel(const float* __restrict__ Q,
                                 const float* __restrict__ Kmat,
                                 float* __restrict__ Attn) {
    const int tid  = threadIdx.x;
    const int wave = tid >> 5;
    const int lane = tid & 31;
    const int col0 = blockIdx.x * 64;
    const int row0 = blockIdx.y * 128 + wave * 16;
    const int bh   = blockIdx.z;             // b*8 + h
    const int bb   = bh >> 3, h = bh & 7;

    const float* qbase = Q    + (size_t)bb * L_SEQ * D_MODEL + h * E_HEAD;
    const float* kbase = Kmat + (size_t)bb * L_SEQ * D_MODEL + h * E_HEAD;

    const int m  = lane & 15;
    const int kh = (lane >> 4) << 1;
    const float* arow = qbase + (size_t)(row0 + m) * D_MODEL;
    const float* b0p  = kbase + (size_t)(col0      + m) * D_MODEL;
    const float* b1p  = kbase + (size_t)(col0 + 16 + m) * D_MODEL;
    const float* b2p  = kbase + (size_t)(col0 + 32 + m) * D_MODEL;
    const float* b3p  = kbase + (size_t)(col0 + 48 + m) * D_MODEL;

    v8f acc0 = {}, acc1 = {}, acc2 = {}, acc3 = {};
    #pragma unroll
    for (int kb = 0; kb < E_HEAD; kb += 4) {
        v2f a  = *(const v2f*)(arow + kb + kh);
        v2f b0 = *(const v2f*)(b0p  + kb + kh);
        v2f b1 = *(const v2f*)(b1p  + kb + kh);
        v2f b2 = *(const v2f*)(b2p  + kb + kh);
        v2f b3 = *(const v2f*)(b3p  + kb + kh);
        acc0 = __builtin_amdgcn_wmma_f32_16x16x4_f32(false, a, false, b0, (short)0, acc0, false, false);
        acc1 = __builtin_amdgcn_wmma_f32_16x16x4_f32(false, a, false, b1, (short)0, acc1, false, false);
        acc2 = __builtin_amdgcn_wmma_f32_16x16x4_f32(false, a, false, b2, (short)0, acc2, false, false);
        acc3 = __builtin_amdgcn_wmma_f32_16x16x4_f32(false, a, false, b3, (short)0, acc3, false, false);
    }

    float* orow = Attn + ((size_t)bh * L_SEQ + row0 + ((lane >> 4) << 3)) * L_SEQ + col0 + m;
    #pragma unroll
    for (int r = 0; r < 8; ++r) {
        orow[(size_t)r * L_SEQ +  0] = acc0[r];
        orow[(size_t)r * L_SEQ + 16] = acc1[r];
        orow[(size_t)r * L_SEQ + 32] = acc2[r];
        orow[(size_t)r * L_SEQ + 48] = acc3[r];
    }
}

// ---------------------------------------------------------------------------
// Kernel 3: attn_i rows are strided row-subsets of attn_1 -> pure row copy.
// One block per destination row; 512 float4 per row.
// ---------------------------------------------------------------------------
__launch_bounds__(256)
__global__ void strided_copy_kernel(const float* __restrict__ src,
                                    float* __restrict__ dst,
                                    int Li, int stride) {
    const int r  = blockIdx.x;          // 0 .. 16*Li-1
    const int bh = r / Li;
    const int lp = r - bh * Li;
    const float4* s = (const float4*)(src + ((size_t)bh * L_SEQ + (size_t)lp * stride) * L_SEQ);
    float4*       d = (float4*)(dst + (size_t)r * L_SEQ);
    d[threadIdx.x]       = s[threadIdx.x];
    d[threadIdx.x + 256] = s[threadIdx.x + 256];
}

// ---------------------------------------------------------------------------
// Kernel 4: per-row softmax stats over attn1: stats[row] = {max, 1/sum(exp)}
// ---------------------------------------------------------------------------
__launch_bounds__(256)
__global__ void row_stats_kernel(const float* __restrict__ Attn,
                                 float* __restrict__ stats) {
    __shared__ float red[256];
    const int row = blockIdx.x;
    const int tid = threadIdx.x;
    const float4* rp = (const float4*)(Attn + (size_t)row * L_SEQ);
    float4 x0 = rp[tid];
    float4 x1 = rp[tid + 256];

    float mx = fmaxf(fmaxf(fmaxf(x0.x, x0.y), fmaxf(x0.z, x0.w)),
                     fmaxf(fmaxf(x1.x, x1.y), fmaxf(x1.z, x1.w)));
    red[tid] = mx; __syncthreads();
    for (int s = 128; s > 0; s >>= 1) {
        if (tid < s) red[tid] = fmaxf(red[tid], red[tid + s]);
        __syncthreads();
    }
    mx = red[0]; __syncthreads();

    float sum = __expf(x0.x - mx) + __expf(x0.y - mx) + __expf(x0.z - mx) + __expf(x0.w - mx)
              + __expf(x1.x - mx) + __expf(x1.y - mx) + __expf(x1.z - mx) + __expf(x1.w - mx);
    red[tid] = sum; __syncthreads();
    for (int s = 128; s > 0; s >>= 1) {
        if (tid < s) red[tid] += red[tid + s];
        __syncthreads();
    }
    if (tid == 0) {
        stats[(size_t)row * 2]     = mx;
        stats[(size_t)row * 2 + 1] = 1.0f / red[0];
    }
}

// ---------------------------------------------------------------------------
// Kernel 5: fused 3-iteration column sum of softmax rows:
//   colsum[bh][s] = sum_l (1 + [l%2==0] + [l%3==0]) * exp(attn1[l,s]-m_l)/sum_l
// Row stats staged in LDS; attn1 columns read fully coalesced (thread = s).
// ---------------------------------------------------------------------------
__launch_bounds__(256)
__global__ void colsum_kernel(const float* __restrict__ Attn,
                              const float* __restrict__ stats,
                              float* __restrict__ colsum) {
    __shared__ float2 st[L_SEQ];   // 16 KB of {max, invsum} for this (b,h)
    const int bh = blockIdx.y;
    const int s0 = blockIdx.x * 256 + threadIdx.x;
    const float2* sg = (const float2*)stats + (size_t)bh * L_SEQ;
    for (int i = threadIdx.x; i < L_SEQ; i += 256) st[i] = sg[i];
    __syncthreads();

    const float* base = Attn + (size_t)bh * L_SEQ * L_SEQ + s0;
    float acc = 0.0f;
    for (int l = 0; l < L_SEQ; ++l) {
        if (l + 8 < L_SEQ)
            __builtin_prefetch(base + (size_t)(l + 8) * L_SEQ, 0, 0);  // global_prefetch_b8
        const float  x  = base[(size_t)l * L_SEQ];
        const float2 ms = st[l];
        const float  w  = 1.0f + (((l & 1) == 0) ? 1.0f : 0.0f)
                               + ((l % 3 == 0)   ? 1.0f : 0.0f);
        acc += w * __expf(x - ms.x) * ms.y;
    }
    colsum[(size_t)bh * L_SEQ + s0] = acc;
}

// ---------------------------------------------------------------------------
// Kernel 6: out[b,s,h*64+e] = v[b,s,h,e] * colsum[b,h,s] / 3
// ---------------------------------------------------------------------------
__launch_bounds__(256)
__global__ void final_out_kernel(const float* __restrict__ V,
                                 const float* __restrict__ colsum,
                                 float* __restrict__ Out) {
    const size_t idx = (size_t)blockIdx.x * 256 + threadIdx.x;
    const int    col = (int)(idx & (D_MODEL - 1));
    const size_t row = idx >> 9;             // b*2048 + s
    const int    b   = (int)(row >> 11);
    const int    s   = (int)(row & (L_SEQ - 1));
    const int    h   = col >> 6;
    Out[idx] = V[idx] * colsum[(((size_t)b * N_HEADS + h) << 11) + s] * (1.0f / 3.0f);
}

// ---------------------------------------------------------------------------
extern "C" void kernel_launch(void* const* d_in, const int* in_sizes, int n_in,
                              void* d_out, int out_size, void* d_ws, size_t ws_size,
                              hipStream_t stream) {
    (void)in_sizes; (void)n_in; (void)out_size; (void)ws_size;

    const float* queries = (const float*)d_in[0];
    const float* keys    = (const float*)d_in[1];
    const float* values  = (const float*)d_in[2];
    const float* Wq      = (const float*)d_in[3];
    const float* bq      = (const float*)d_in[4];
    const float* Wk      = (const float*)d_in[5];
    const float* bk      = (const float*)d_in[6];
    const float* Wv      = (const float*)d_in[7];
    const float* bv      = (const float*)d_in[8];

    float* out = (float*)d_out;
    float* ws  = (float*)d_ws;

    // workspace layout (floats): q,k,v (2,097,152 each), row stats, colsum (~25.6 MB)
    float* q     = ws;
    float* k     = ws + 2097152;
    float* v     = ws + 4194304;
    float* stats = ws + 6291456;   // 32768 rows x {max, invsum}
    float* csum  = ws + 6356992;   // 16 x 2048

    // output layout: [out | attn1 | attn2 | attn3]
    float* out_final = out;
    float* attn1 = out + 2097152;             // 16 x 2048 x 2048
    float* attn2 = out + 69206016;            // 16 x 1024 x 2048
    float* attn3 = out + 102760448;           // 16 x  683 x 2048

    const dim3 pgrid(D_MODEL / 16, 4096 / 128);   // (32, 32), 8 waves/block
    qkv_proj_kernel<<<pgrid, 256, 0, stream>>>(queries, Wq, bq, q);
    qkv_proj_kernel<<<pgrid, 256, 0, stream>>>(keys,    Wk, bk, k);
    qkv_proj_kernel<<<pgrid, 256, 0, stream>>>(values,  Wv, bv, v);

    attn_gemm_kernel<<<dim3(L_SEQ / 64, L_SEQ / 128, 16), 256, 0, stream>>>(q, k, attn1);

    strided_copy_kernel<<<16 * 1024, 256, 0, stream>>>(attn1, attn2, 1024, 2);
    strided_copy_kernel<<<16 * 683,  256, 0, stream>>>(attn1, attn3,  683, 3);

    row_stats_kernel<<<16 * L_SEQ, 256, 0, stream>>>(attn1, stats);
    colsum_kernel<<<dim3(L_SEQ / 256, 16), 256, 0, stream>>>(attn1, stats, csum);
    final_out_kernel<<<2097152 / 256, 256, 0, stream>>>(v, csum, out_final);
}